// GCN_34591666602590
// MI455X (gfx1250) — compile-verified
//
#include <hip/hip_runtime.h>
#include <hip/hip_bf16.h>

// ---------------------------------------------------------------------------
// GCN on MI455X (gfx1250, wave32).
// Pipeline:
//   deg/CSR build -> h1 = nodes@W1 (WMMA f32 16x16x4)
//   x1 = CSR-gather(h1)*norm + b1            (relu deferred to GEMM2 A-load)
//   h2 = relu(x1)@W2 (WMMA)
//   x2 = CSR-gather(h2)*norm + b2            (relu deferred to matvec x-load)
//   y1 = relu(x2_flat) @ Wl1  (600 MB NT stream -> the roofline limiter)
//   out = sigmoid(relu(y1+bl1) @ Wl2 + bl2)
// ---------------------------------------------------------------------------

typedef float v2f __attribute__((ext_vector_type(2)));
typedef float v8f __attribute__((ext_vector_type(8)));

#define F_IN 100
#define H1   200
#define H2   150
#define HID  100
#define NOUT 12

// ---------------- small utility kernels ----------------
__global__ void zero_i32_kernel(int* __restrict__ p, int n) {
  int i = blockIdx.x * blockDim.x + threadIdx.x;
  if (i < n) p[i] = 0;
}

__global__ void zero_f32_kernel(float* __restrict__ p, int n) {
  int i = blockIdx.x * blockDim.x + threadIdx.x;
  if (i < n) p[i] = 0.0f;
}

__global__ void count_edges_kernel(const int* __restrict__ dst, int E,
                                   int* __restrict__ cnt) {
  int i = blockIdx.x * blockDim.x + threadIdx.x;
  if (i < E) atomicAdd(&cnt[dst[i]], 1);
}

// deg includes the self loop: deg = cnt + 1  (always >= 1)
__global__ void dinv_kernel(const int* __restrict__ cnt, float* __restrict__ dinv,
                            int n) {
  int i = blockIdx.x * blockDim.x + threadIdx.x;
  if (i < n) dinv[i] = rsqrtf((float)(cnt[i] + 1));
}

// Single-block exclusive scan over n counts -> row_ptr[0..n]
__global__ __launch_bounds__(1024) void scan_kernel(const int* __restrict__ cnt,
                                                    int* __restrict__ row_ptr,
                                                    int n) {
  __shared__ int sh[1024];
  __shared__ int carry_s;
  const int tid = threadIdx.x;
  if (tid == 0) carry_s = 0;
  __syncthreads();
  for (int base = 0; base < n; base += 1024) {
    int i = base + tid;
    int v = (i < n) ? cnt[i] : 0;
    sh[tid] = v;
    __syncthreads();
    // Hillis-Steele inclusive scan
    for (int off = 1; off < 1024; off <<= 1) {
      int t = (tid >= off) ? sh[tid - off] : 0;
      __syncthreads();
      sh[tid] += t;
      __syncthreads();
    }
    int c = carry_s;
    if (i < n) row_ptr[i] = c + sh[tid] - v;   // exclusive
    int total = sh[1023];
    __syncthreads();
    if (tid == 0) carry_s = c + total;
    __syncthreads();
  }
  if (tid == 0) row_ptr[n] = carry_s;
}

__global__ void copy_i32_kernel(const int* __restrict__ s, int* __restrict__ d,
                                int n) {
  int i = blockIdx.x * blockDim.x + threadIdx.x;
  if (i < n) d[i] = s[i];
}

__global__ void fill_csr_kernel(const int* __restrict__ esrc,
                                const int* __restrict__ edst, int E,
                                int* __restrict__ fill, int* __restrict__ col) {
  int i = blockIdx.x * blockDim.x + threadIdx.x;
  if (i < E) {
    int p = atomicAdd(&fill[edst[i]], 1);
    col[p] = esrc[i];
  }
}

// ---------------- WMMA GEMM: C[M,Nc] = (relu?)A[M,K] @ B[K,Nc] ----------------
// One wave per 16x16 output tile, V_WMMA_F32_16X16X4_F32, K multiple of 4.
// A layout per lane (16x4 f32): m = lane&15, ks = k0 + (lane>>4)*2, regs {ks,ks+1}
// B layout per lane (4x16 f32): n = lane&15, same k split
// C/D layout: n = lane&15, VGPR v holds m = v + (lane>>4)*8
//
// Divergence-free inner loop: load addresses are CLAMPED in-bounds (garbage in
// clamped columns/rows only feeds D entries that are masked at the store, since
// D[:,n] depends only on B[:,n] and D[m,:] only on A[m,:]). EXEC stays all-1s
// for every v_wmma as the ISA requires.
template <bool RELU_A>
__global__ __launch_bounds__(256) void wmma_gemm_f32(const float* __restrict__ A,
                                                     const float* __restrict__ B,
                                                     float* __restrict__ C,
                                                     int M, int K, int Nc) {
  const int lane = threadIdx.x & 31;
  const int wave = threadIdx.x >> 5;
  const int tileCols = (Nc + 15) >> 4;
  const int tileRows = (M + 15) >> 4;
  int tile = blockIdx.x * (blockDim.x >> 5) + wave;
  if (tile >= tileRows * tileCols) return;   // uniform per wave: EXEC stays full
  const int tm = tile / tileCols;
  const int tn = tile % tileCols;

  const int mrow  = tm * 16 + (lane & 15);
  const int ncol  = tn * 16 + (lane & 15);
  const int khalf = (lane >> 4) * 2;
  const int mrowC = mrow < M ? mrow : (M - 1);      // clamp, no predication
  const int ncolC = ncol < Nc ? ncol : (Nc - 1);

  const float* __restrict__ Ap = A + (long)mrowC * K + khalf;   // 8B-aligned
  const float* __restrict__ Bp = B + (long)khalf * Nc + ncolC;
  const long bstep = Nc;

  v8f c = {};
#pragma unroll 5
  for (int k0 = 0; k0 < K; k0 += 4) {
    v2f a = *(const v2f*)(Ap + k0);                 // global_load_b64
    if (RELU_A) { a[0] = fmaxf(a[0], 0.0f); a[1] = fmaxf(a[1], 0.0f); }
    v2f b;
    b[0] = Bp[(long)k0 * bstep];                    // row k0+khalf
    b[1] = Bp[(long)k0 * bstep + bstep];            // row k0+khalf+1
    c = __builtin_amdgcn_wmma_f32_16x16x4_f32(false, a, false, b, (short)0, c,
                                              false, false);
  }
  if (ncol < Nc) {
    const int mbase = tm * 16 + (lane >> 4) * 8;
#pragma unroll
    for (int v = 0; v < 8; ++v) {
      int m = mbase + v;
      if (m < M) C[(long)m * Nc + ncol] = c[v];
    }
  }
}

// ---------------- CSR gather aggregation (GCNConv message passing) ----------
// out[i,f] = b[f] + h[i,f]*dinv[i]^2 + sum_{e: dst=i} h[src,f]*dinv[src]*dinv[i]
// One block per node; lanes own features (coalesced h-row reads from L2).
// No barriers -> lanes beyond F may simply retire (EXEC set once, no per-edge
// re-predication).
__global__ void aggregate_kernel(const float* __restrict__ h,
                                 const int* __restrict__ row_ptr,
                                 const int* __restrict__ col,
                                 const float* __restrict__ dinv,
                                 const float* __restrict__ bias,
                                 float* __restrict__ out, int F) {
  const int node = blockIdx.x;
  const int f = threadIdx.x;
  if (f >= F) return;
  const float di = dinv[node];
  const int beg = row_ptr[node];
  const int end = row_ptr[node + 1];
  float acc = h[(long)node * F + f] * di * di + bias[f];
  for (int e = beg; e < end; ++e) {
    const int s = col[e];                    // broadcast load
    const float w = dinv[s] * di;
    acc = fmaf(h[(long)s * F + f], w, acc);
  }
  out[(long)node * F + f] = acc;
}

// ---------------- flatten matvec: y1[j] += relu(x[i]) * Wl1[i,j] -------------
// 600 MB Wl1 streamed once with non-temporal loads (protect L2-resident graph
// tensors). 128-lane groups: j = tid&127 (j<100 active), rows grid-strided.
__global__ __launch_bounds__(1024) void matvec_nt_kernel(
    const float* __restrict__ x, const float* __restrict__ Wl1,
    float* __restrict__ y1, long rows) {
  const int tid = threadIdx.x;
  const int j = tid & 127;
  const int g = tid >> 7;                       // 8 groups / block
  const long group = (long)blockIdx.x * 8 + g;
  const long nGroups = (long)gridDim.x * 8;
  float acc = 0.0f;
  if (j < HID) {
    for (long i = group; i < rows; i += nGroups) {
      const float xv = fmaxf(x[i], 0.0f);       // deferred relu of layer 2
      const float w = __builtin_nontemporal_load(&Wl1[i * HID + j]);
      acc = fmaf(xv, w, acc);
    }
  }
  __shared__ float red[1024];
  red[tid] = acc;
  __syncthreads();
#pragma unroll
  for (int s = 4; s >= 1; s >>= 1) {            // reduce the 8 groups
    if (g < s) red[tid] += red[tid + s * 128];
    __syncthreads();
  }
  if (tid < HID) atomicAdd(&y1[tid], red[tid]);
}

// ---------------- MLP head: out = sigmoid(relu(y1+bl1) @ Wl2 + bl2) ----------
__global__ void head_kernel(const float* __restrict__ y1,
                            const float* __restrict__ bl1,
                            const float* __restrict__ Wl2,
                            const float* __restrict__ bl2,
                            float* __restrict__ out) {
  __shared__ float t[HID];
  const int tid = threadIdx.x;
  if (tid < HID) t[tid] = fmaxf(y1[tid] + bl1[tid], 0.0f);
  __syncthreads();
  if (tid < NOUT) {
    float z = bl2[tid];
    for (int k = 0; k < HID; ++k) z = fmaf(t[k], Wl2[k * NOUT + tid], z);
    out[tid] = 1.0f / (1.0f + expf(-z));
  }
}

// ---------------------------------------------------------------------------
static inline char* carve(char*& p, size_t bytes) {
  char* r = p;
  p += (bytes + 255) & ~size_t(255);
  return r;
}

extern "C" void kernel_launch(void* const* d_in, const int* in_sizes, int n_in,
                              void* d_out, int out_size, void* d_ws,
                              size_t ws_size, hipStream_t stream) {
  const float* nodes = (const float*)d_in[0];
  const int*   edges = (const int*)d_in[1];
  const float* W1  = (const float*)d_in[2];
  const float* b1  = (const float*)d_in[3];
  const float* W2  = (const float*)d_in[4];
  const float* b2  = (const float*)d_in[5];
  const float* Wl1 = (const float*)d_in[6];
  const float* bl1 = (const float*)d_in[7];
  const float* Wl2 = (const float*)d_in[8];
  const float* bl2 = (const float*)d_in[9];
  float* out = (float*)d_out;

  const int N = in_sizes[0] / F_IN;   // 10000
  const int E = in_sizes[1] / 2;      // 320000
  const int* esrc = edges;            // edges[0,:]
  const int* edst = edges + E;        // edges[1,:]

  // workspace carve (~30 MB)
  char* p = (char*)d_ws;
  int*   cnt     = (int*)  carve(p, sizeof(int) * N);
  int*   row_ptr = (int*)  carve(p, sizeof(int) * (N + 1));
  int*   fill    = (int*)  carve(p, sizeof(int) * N);
  int*   col     = (int*)  carve(p, sizeof(int) * E);
  float* dinv    = (float*)carve(p, sizeof(float) * N);
  float* h1      = (float*)carve(p, sizeof(float) * (size_t)N * H1);
  float* x1      = (float*)carve(p, sizeof(float) * (size_t)N * H1);
  float* h2      = (float*)carve(p, sizeof(float) * (size_t)N * H2);
  float* x2      = (float*)carve(p, sizeof(float) * (size_t)N * H2);
  float* y1      = (float*)carve(p, sizeof(float) * 128);
  (void)ws_size; (void)n_in; (void)out_size;

  // --- degree + CSR build ---
  zero_i32_kernel<<<(N + 255) / 256, 256, 0, stream>>>(cnt, N);
  count_edges_kernel<<<(E + 255) / 256, 256, 0, stream>>>(edst, E, cnt);
  dinv_kernel<<<(N + 255) / 256, 256, 0, stream>>>(cnt, dinv, N);
  scan_kernel<<<1, 1024, 0, stream>>>(cnt, row_ptr, N);
  copy_i32_kernel<<<(N + 255) / 256, 256, 0, stream>>>(row_ptr, fill, N);
  fill_csr_kernel<<<(E + 255) / 256, 256, 0, stream>>>(esrc, edst, E, fill, col);

  // --- layer 1: GEMM + aggregate ---
  {
    const int tiles = ((N + 15) / 16) * ((H1 + 15) / 16);
    wmma_gemm_f32<false><<<(tiles + 7) / 8, 256, 0, stream>>>(nodes, W1, h1,
                                                              N, F_IN, H1);
  }
  aggregate_kernel<<<N, 224, 0, stream>>>(h1, row_ptr, col, dinv, b1, x1, H1);

  // --- layer 2: GEMM (relu fused on A) + aggregate ---
  {
    const int tiles = ((N + 15) / 16) * ((H2 + 15) / 16);
    wmma_gemm_f32<true><<<(tiles + 7) / 8, 256, 0, stream>>>(x1, W2, h2,
                                                             N, H1, H2);
  }
  aggregate_kernel<<<N, 160, 0, stream>>>(h2, row_ptr, col, dinv, b2, x2, H2);

  // --- flatten matvec over 600 MB Wl1 (relu fused on x) ---
  zero_f32_kernel<<<1, 128, 0, stream>>>(y1, 128);
  matvec_nt_kernel<<<512, 1024, 0, stream>>>(x2, Wl1, y1, (long)N * H2);

  // --- head ---
  head_kernel<<<1, 128, 0, stream>>>(y1, bl1, Wl2, bl2, out);
}